// Get_Cat_Feat_Tgt_29540785062245
// MI455X (gfx1250) — compile-verified
//
#include <hip/hip_runtime.h>
#include <stdint.h>

typedef __attribute__((ext_vector_type(2))) float v2f;
typedef __attribute__((ext_vector_type(8))) float v8f;

#define BATCH    2
#define M_PTS    4096          // 64*64 candidate points per batch
#define N_TGT    16384
#define F_DIM    32
#define NSAMPLE  32
#define KNN_K    32
#define OUTW     35            // 3 xyz + 32 feat
#define RADIUS2  1.0f

#define TILE_M   16
#define NCHUNK   512           // targets staged per LDS chunk

// ---------------- helpers ----------------

__device__ __forceinline__ unsigned f2u_mono(float f) {
  unsigned b = __float_as_uint(f);
  // monotone map: order of mapped uints == order of floats (handles negatives)
  return b ^ (unsigned)(((int)b >> 31) | 0x80000000);
}
__device__ __forceinline__ float u2f_mono(unsigned u) {
  unsigned b = (u & 0x80000000u) ? (u ^ 0x80000000u) : ~u;
  return __uint_as_float(b);
}

__device__ __forceinline__ unsigned long long shfl_xor_u64(unsigned long long v, int m) {
  unsigned lo = __shfl_xor((unsigned)v, m, 32);
  unsigned hi = __shfl_xor((unsigned)(v >> 32), m, 32);
  return ((unsigned long long)hi << 32) | lo;
}
__device__ __forceinline__ unsigned long long shfl_u64(unsigned long long v, int src) {
  unsigned lo = __shfl((unsigned)v, src, 32);
  unsigned hi = __shfl((unsigned)(v >> 32), src, 32);
  return ((unsigned long long)hi << 32) | lo;
}

// full bitonic sort of 32 keys across the wave (ascending by lane)
__device__ __forceinline__ unsigned long long bitonic_sort32(unsigned long long key, int lane) {
  #pragma unroll
  for (int k = 2; k <= 32; k <<= 1) {
    #pragma unroll
    for (int j = k >> 1; j > 0; j >>= 1) {
      unsigned long long v = shfl_xor_u64(key, j);
      bool up     = ((lane & k) == 0);
      bool lower  = ((lane & j) == 0);
      unsigned long long mn = (v < key) ? v : key;
      unsigned long long mx = (v < key) ? key : v;
      key = (up == lower) ? mn : mx;
    }
  }
  return key;
}

// merge a bitonic sequence into ascending order
__device__ __forceinline__ unsigned long long bitonic_merge32(unsigned long long key, int lane) {
  #pragma unroll
  for (int j = 16; j > 0; j >>= 1) {
    unsigned long long v = shfl_xor_u64(key, j);
    unsigned long long mn = (v < key) ? v : key;
    unsigned long long mx = (v < key) ? key : v;
    key = ((lane & j) == 0) ? mn : mx;
  }
  return key;
}

// ---------------- kernel 1: radius grouping (out cols 0..2) ----------------

__global__ __launch_bounds__(256) void group_xyz_kernel(const float* __restrict__ cand_pts,
                                                        float* __restrict__ out) {
  __shared__ int s_g[8][NSAMPLE];
  const int tid  = threadIdx.x;
  const int lane = tid & 31;
  const int wv   = tid >> 5;
  const int rowg = blockIdx.x * 8 + wv;           // 0 .. B*M-1
  const int b = rowg / M_PTS;
  const int m = rowg % M_PTS;

  const float* pb = cand_pts + (size_t)b * M_PTS * 3;
  const float mx = pb[m * 3 + 0], my = pb[m * 3 + 1], mz = pb[m * 3 + 2];
  const float mn2 = mx * mx + my * my + mz * mz;

  int cnt = 0;
  for (int n0 = 0; n0 < M_PTS && cnt < NSAMPLE; n0 += 32) {
    int n = n0 + lane;
    float x = pb[n * 3 + 0], y = pb[n * 3 + 1], z = pb[n * 3 + 2];
    float bn2 = x * x + y * y + z * z;
    float d2  = mn2 + bn2 - 2.0f * (mx * x + my * y + mz * z);  // same expansion as reference
    bool q = (d2 <= RADIUS2);
    unsigned mask = (unsigned)__ballot(q);
    int pos = cnt + __popc(mask & ((1u << lane) - 1u));
    if (q && pos < NSAMPLE) s_g[wv][pos] = n;
    cnt += __popc(mask);
  }
  __syncthreads();

  int total = cnt < NSAMPLE ? cnt : NSAMPLE;      // cnt >= 1 always (self qualifies)
  int g0 = s_g[wv][0];
  int gi = (lane < total) ? s_g[wv][lane] : g0;   // pad with first index, like reference

  size_t ob = (((size_t)b * M_PTS + m) * NSAMPLE + lane) * OUTW;
  out[ob + 0] = pb[gi * 3 + 0] - mx;
  out[ob + 1] = pb[gi * 3 + 1] - my;
  out[ob + 2] = pb[gi * 3 + 2] - mz;
}

// ---------------- kernel 2: KNN + weighted features (out cols 3..34) ----------------

__global__ __launch_bounds__(256) void knn_feat_kernel(const float* __restrict__ cand_pts,
                                                       const float* __restrict__ tgt_xyz,
                                                       const float* __restrict__ tgt_feat,
                                                       float* __restrict__ out) {
  __shared__ float s_d2[TILE_M][NCHUNK];                      // 32 KB
  __shared__ float s_bx[NCHUNK], s_by[NCHUNK], s_bz[NCHUNK], s_bn2[NCHUNK];  // 8 KB
  __shared__ float s_ax[TILE_M], s_ay[TILE_M], s_az[TILE_M], s_an2[TILE_M];

  const int tid   = threadIdx.x;
  const int lane  = tid & 31;
  const int wv    = tid >> 5;                                 // 0..7
  const bool lo16 = (lane < 16);
  const int b     = blockIdx.x / (M_PTS / TILE_M);
  const int m0    = (blockIdx.x % (M_PTS / TILE_M)) * TILE_M;

  if (tid < TILE_M) {
    float x = cand_pts[((size_t)b * M_PTS + m0 + tid) * 3 + 0];
    float y = cand_pts[((size_t)b * M_PTS + m0 + tid) * 3 + 1];
    float z = cand_pts[((size_t)b * M_PTS + m0 + tid) * 3 + 2];
    s_ax[tid] = x; s_ay[tid] = y; s_az[tid] = z;
    s_an2[tid] = x * x + y * y + z * z;
  }
  __syncthreads();

  // A operand (16x4 f32, pre-scaled by -2 so WMMA emits -2*dot).
  // Layout (ISA 16x4 f32 A): lanes 0-15: M=lane, K0/K1 = x/y; lanes 16-31: M=lane-16, K2/K3 = z/0.
  // Built branchlessly (cndmask instead of exec branches).
  v2f a;
  {
    int mr = lane & 15;
    float ax = s_ax[mr], ay = s_ay[mr], az = s_az[mr];
    a.x = -2.0f * (lo16 ? ax : az);
    a.y = lo16 ? (-2.0f * ay) : 0.0f;
  }
  const int mhalf = 8 * (lane >> 4);

  // top-32 running state for rows wv*2 and wv*2+1, plus cached 32nd-best key
  unsigned long long run0 = ~0ULL, run1 = ~0ULL;
  unsigned long long runMax0 = ~0ULL, runMax1 = ~0ULL;
  const float* tb = tgt_xyz + (size_t)b * N_TGT * 3;

  for (int n0 = 0; n0 < N_TGT; n0 += NCHUNK) {
    // stage chunk coords + squared norms (coalesced)
    for (int p = tid; p < NCHUNK; p += 256) {
      float x = tb[(n0 + p) * 3 + 0];
      float y = tb[(n0 + p) * 3 + 1];
      float z = tb[(n0 + p) * 3 + 2];
      s_bx[p] = x; s_by[p] = y; s_bz[p] = z;
      s_bn2[p] = x * x + y * y + z * z;
    }
    __syncthreads();

    // d2 tile via WMMA: 32 n-tiles, 4 per wave.  C preloaded with |a|^2+|b|^2.
    for (int t = wv; t < NCHUNK / 16; t += 8) {
      int nc = t * 16 + (lane & 15);
      float bx = s_bx[nc], by = s_by[nc], bz = s_bz[nc];
      v2f bv;                               // B 4x16 f32: lanes 0-15 rows K0/K1, lanes 16-31 rows K2/K3
      bv.x = lo16 ? bx : bz;
      bv.y = lo16 ? by : 0.0f;
      v8f c;
      float bn2 = s_bn2[nc];
      #pragma unroll
      for (int r = 0; r < 8; ++r) c[r] = s_an2[r + mhalf] + bn2;
      c = __builtin_amdgcn_wmma_f32_16x16x4_f32(false, a, false, bv, (short)0, c, false, false);
      #pragma unroll
      for (int r = 0; r < 8; ++r) s_d2[r + mhalf][nc] = c[r]; // exact fp32 d2
    }
    __syncthreads();

    // streaming top-32: 2 rows per wave, cached runMax for the early-skip test
    #pragma unroll
    for (int rr = 0; rr < 2; ++rr) {
      int row = wv * 2 + rr;
      unsigned long long run    = rr ? run1 : run0;
      unsigned long long runMax = rr ? runMax1 : runMax0;
      for (int sc = 0; sc < NCHUNK / 32; ++sc) {
        int nn = sc * 32 + lane;
        float d2 = s_d2[row][nn];
        unsigned long long key =
            ((unsigned long long)f2u_mono(d2) << 32) | (unsigned)(n0 + nn);
        if (__ballot(key < runMax) != 0) {                    // uniform, usually false once warm
          key = bitonic_sort32(key, lane);
          unsigned long long rev = shfl_xor_u64(key, 31);     // descending
          unsigned long long mrg = (run < rev) ? run : rev;   // lowest 32, bitonic
          run = bitonic_merge32(mrg, lane);
          runMax = shfl_u64(run, 31);
        }
      }
      if (rr) { run1 = run; runMax1 = runMax; }
      else    { run0 = run; runMax0 = runMax; }
    }
    __syncthreads();
  }

  // finalize: weights + feature gather.  lane == sample slot s (sorted asc by (d2, idx))
  #pragma unroll
  for (int rr = 0; rr < 2; ++rr) {
    int row = wv * 2 + rr;
    unsigned long long key = rr ? run1 : run0;
    unsigned idx = (unsigned)(key & 0xFFFFFFFFu);
    float d2 = u2f_mono((unsigned)(key >> 32));
    float dist = sqrtf(fmaxf(d2, 0.0f));
    float s = dist;
    #pragma unroll
    for (int j = 16; j > 0; j >>= 1) s += __shfl_xor(s, j, 32);
    float w = dist / s;

    const float* fp = tgt_feat + ((size_t)b * N_TGT + idx) * F_DIM;
    size_t ob = (((size_t)b * M_PTS + m0 + row) * NSAMPLE + lane) * OUTW;
    #pragma unroll
    for (int f = 0; f < F_DIM; f += 4) {
      float4 v = *(const float4*)(fp + f);
      out[ob + 3 + f + 0] = v.x * w;
      out[ob + 3 + f + 1] = v.y * w;
      out[ob + 3 + f + 2] = v.z * w;
      out[ob + 3 + f + 3] = v.w * w;
    }
  }
}

// ---------------- launch ----------------

extern "C" void kernel_launch(void* const* d_in, const int* in_sizes, int n_in,
                              void* d_out, int out_size, void* d_ws, size_t ws_size,
                              hipStream_t stream) {
  const float* cand_pts = (const float*)d_in[0];
  // d_in[1] (src_keypts) is unused by the reference
  const float* tgt_xyz  = (const float*)d_in[2];
  const float* tgt_feat = (const float*)d_in[3];
  float* out = (float*)d_out;

  // kernel 1: 8192 rows, one wave each, 8 waves per block
  group_xyz_kernel<<<(BATCH * M_PTS) / 8, 256, 0, stream>>>(cand_pts, out);
  // kernel 2: one block per 16-row tile
  knn_feat_kernel<<<BATCH * (M_PTS / TILE_M), 256, 0, stream>>>(cand_pts, tgt_xyz, tgt_feat, out);
}